// OSELM_68324339745369
// MI455X (gfx1250) — compile-verified
//
#include <hip/hip_runtime.h>
#include <hip/hip_bf16.h>

typedef __attribute__((ext_vector_type(2))) float v2f;
typedef __attribute__((ext_vector_type(8))) float v8f;
typedef int v4i __attribute__((vector_size(16)));  // pointee type for async-LDS builtins

#if defined(__has_builtin)
#if __has_builtin(__builtin_amdgcn_global_load_async_to_lds_b128) && \
    __has_builtin(__builtin_amdgcn_s_wait_asynccnt)
#define USE_ASYNC_LDS 1
#else
#define USE_ASYNC_LDS 0
#endif
#else
#define USE_ASYNC_LDS 0
#endif

#define BM 64
#define BN 64
#define BKK 32
// LDS strides (floats). Chosen so 16B stores stay aligned and the strided
// per-lane fragment reads hit distinct banks (64 banks x 4B).
#define SA_N 36  // !TA: As[m*SA_N + k]
#define SA_T 68  //  TA: As[k*SA_T + m]
#define SB_N 68  // !TB: Bs[k*SB_N + n]
#define SB_T 36  //  TB: Bs[n*SB_T + k]

// C[MxN] = alpha * op(A)op(B) + betac * D + gammad * I
// op(A): !TA -> A is [M,K] row-major;  TA -> A is [K,M] row-major (A^T)
// op(B): !TB -> B is [K,N];            TB -> B is [N,K] (B^T)
// Requires M,N multiples of 64 and K multiple of 32. D uses ldc.
template <bool TA, bool TB>
__global__ __launch_bounds__(128) void gemm_f32_wmma(
    const float* __restrict__ A, const float* __restrict__ B,
    float* __restrict__ C, const float* __restrict__ D,
    int M, int N, int K, int lda, int ldb, int ldc,
    float alpha, float betac, float gammad) {
  __shared__ float As[2][2304];
  __shared__ float Bs[2][2304];

  const int tid  = threadIdx.x;
  const int lane = tid & 31;
  const int wave = tid >> 5;
  const int wm   = (wave >> 1) * 32;  // wave tile origin in block tile
  const int wn   = (wave & 1) * 32;
  const int hf   = lane >> 4;  // half-wave select (K pair / M+8)
  const int l    = lane & 15;

  const int bm = blockIdx.y * BM;
  const int bn = blockIdx.x * BN;

  // ---- per-thread tile-copy descriptors (computed once) ----
  int agoff[4], aldso[4], bgoff[4], bldso[4];
#pragma unroll
  for (int i = 0; i < 4; ++i) {
    int t = tid + i * 128;
    if (!TA) {  // A tile: 64 rows x 8 float4 along K
      int r = t >> 3, c4 = t & 7;
      agoff[i] = (bm + r) * lda + c4 * 4;
      aldso[i] = r * SA_N + c4 * 4;
    } else {  // A is [K,M]: 32 rows (k) x 16 float4 along M
      int r = t >> 4, c4 = t & 15;
      agoff[i] = r * lda + bm + c4 * 4;
      aldso[i] = r * SA_T + c4 * 4;
    }
    if (!TB) {  // B tile: 32 rows (k) x 16 float4 along N
      int r = t >> 4, c4 = t & 15;
      bgoff[i] = r * ldb + bn + c4 * 4;
      bldso[i] = r * SB_N + c4 * 4;
    } else {  // B is [N,K]: 64 rows (n) x 8 float4 along K
      int r = t >> 3, c4 = t & 7;
      bgoff[i] = (bn + r) * ldb + c4 * 4;
      bldso[i] = r * SB_T + c4 * 4;
    }
  }
  const int astep = TA ? BKK * lda : BKK;
  const int bstep = TB ? BKK : BKK * ldb;
  const int nkb   = K / BKK;

  v8f acc[2][2] = {};

  // ---- compute on one LDS buffer: 8 k-steps of 16x16x4 f32 WMMA ----
  auto compute = [&](const float* as, const float* bs) {
#pragma unroll
    for (int kk = 0; kk < BKK; kk += 4) {
      const int ka = kk + 2 * hf;  // lanes 0-15: K=kk,kk+1 ; 16-31: kk+2,kk+3
      v2f a[2], b[2];
#pragma unroll
      for (int tm = 0; tm < 2; ++tm) {
        int m = wm + tm * 16 + l;
        a[tm][0] = TA ? as[(ka + 0) * SA_T + m] : as[m * SA_N + (ka + 0)];
        a[tm][1] = TA ? as[(ka + 1) * SA_T + m] : as[m * SA_N + (ka + 1)];
      }
#pragma unroll
      for (int tn = 0; tn < 2; ++tn) {
        int n = wn + tn * 16 + l;
        b[tn][0] = TB ? bs[n * SB_T + (ka + 0)] : bs[(ka + 0) * SB_N + n];
        b[tn][1] = TB ? bs[n * SB_T + (ka + 1)] : bs[(ka + 1) * SB_N + n];
      }
#pragma unroll
      for (int tm = 0; tm < 2; ++tm)
#pragma unroll
        for (int tn = 0; tn < 2; ++tn)
          acc[tm][tn] = __builtin_amdgcn_wmma_f32_16x16x4_f32(
              false, a[tm], false, b[tn], (short)0, acc[tm][tn], false, false);
    }
  };

#if USE_ASYNC_LDS
  // ---- double-buffered pipeline with async global->LDS (ASYNCcnt) ----
#pragma unroll
  for (int i = 0; i < 4; ++i) {
    __builtin_amdgcn_global_load_async_to_lds_b128(
        (v4i*)(A + agoff[i]), (v4i*)&As[0][aldso[i]], 0, 0);
    __builtin_amdgcn_global_load_async_to_lds_b128(
        (v4i*)(B + bgoff[i]), (v4i*)&Bs[0][bldso[i]], 0, 0);
  }
  __builtin_amdgcn_s_wait_asynccnt(0);
  __syncthreads();
  int aoff = astep, boff = bstep;
  for (int kb = 0; kb < nkb; ++kb) {
    const int buf = kb & 1;
    if (kb + 1 < nkb) {  // prefetch next tile into the other buffer
#pragma unroll
      for (int i = 0; i < 4; ++i) {
        __builtin_amdgcn_global_load_async_to_lds_b128(
            (v4i*)(A + agoff[i] + aoff), (v4i*)&As[buf ^ 1][aldso[i]], 0, 0);
        __builtin_amdgcn_global_load_async_to_lds_b128(
            (v4i*)(B + bgoff[i] + boff), (v4i*)&Bs[buf ^ 1][bldso[i]], 0, 0);
      }
      aoff += astep;
      boff += bstep;
    }
    compute(As[buf], Bs[buf]);
    __builtin_amdgcn_s_wait_asynccnt(0);
    __syncthreads();
  }
#else
  // ---- fallback: register-staged double buffer ----
  float4 ra[4], rb[4];
  auto gload = [&](int ao, int bo) {
#pragma unroll
    for (int i = 0; i < 4; ++i) {
      ra[i] = *reinterpret_cast<const float4*>(A + agoff[i] + ao);
      rb[i] = *reinterpret_cast<const float4*>(B + bgoff[i] + bo);
    }
  };
  auto lstore = [&](int buf) {
#pragma unroll
    for (int i = 0; i < 4; ++i) {
      *reinterpret_cast<float4*>(&As[buf][aldso[i]]) = ra[i];
      *reinterpret_cast<float4*>(&Bs[buf][bldso[i]]) = rb[i];
    }
  };
  gload(0, 0);
  lstore(0);
  __syncthreads();
  int aoff = astep, boff = bstep;
  for (int kb = 0; kb < nkb; ++kb) {
    const int buf = kb & 1;
    const bool pre = (kb + 1 < nkb);
    if (pre) {  // issue next tile's global loads; overlap with compute
      gload(aoff, boff);
      aoff += astep;
      boff += bstep;
    }
    compute(As[buf], Bs[buf]);
    if (pre) lstore(buf ^ 1);
    __syncthreads();
  }
#endif

  // ---- epilogue: C = alpha*acc + betac*D + gammad*I ----
#pragma unroll
  for (int tm = 0; tm < 2; ++tm) {
#pragma unroll
    for (int tn = 0; tn < 2; ++tn) {
#pragma unroll
      for (int i = 0; i < 8; ++i) {
        int row = bm + wm + tm * 16 + hf * 8 + i;  // VGPR i -> M = i + 8*half
        int col = bn + wn + tn * 16 + l;
        float v = alpha * acc[tm][tn][i];
        if (D) v += betac * D[(size_t)row * ldc + col];
        if (gammad != 0.0f && row == col) v += gammad;
        C[(size_t)row * ldc + col] = v;
      }
    }
  }
}

// inv-infinity-norm of an n x n matrix (n <= 512), one block of 512 threads
__global__ __launch_bounds__(512) void rownorm_inf_inv(
    const float* __restrict__ S, int n, float* __restrict__ out) {
  __shared__ float red[512];
  int r = threadIdx.x;
  float s = 0.0f;
  if (r < n) {
    const float* row = S + (size_t)r * n;
    for (int j = 0; j < n; ++j) s += fabsf(row[j]);
  }
  red[r] = s;
  __syncthreads();
  for (int off = 256; off > 0; off >>= 1) {
    if (r < off) red[r] = fmaxf(red[r], red[r + off]);
    __syncthreads();
  }
  if (r == 0) out[0] = 1.0f / red[0];
}

__global__ void init_scaled_identity(float* __restrict__ X, int n,
                                     const float* __restrict__ scale) {
  int idx = blockIdx.x * blockDim.x + threadIdx.x;
  if (idx < n * n) {
    int r = idx / n, c = idx - r * n;
    X[idx] = (r == c) ? scale[0] : 0.0f;
  }
}

// bnew = beta + pG - Z  (elementwise)
__global__ void beta_combine(const float* __restrict__ beta,
                             const float* __restrict__ pG,
                             const float* __restrict__ Z,
                             float* __restrict__ out, int n) {
  int i = blockIdx.x * blockDim.x + threadIdx.x;
  if (i < n) out[i] = beta[i] + pG[i] - Z[i];
}

extern "C" void kernel_launch(void* const* d_in, const int* in_sizes, int n_in,
                              void* d_out, int out_size, void* d_ws, size_t ws_size,
                              hipStream_t stream) {
  constexpr int Bz = 512, Nh = 4096, No = 128;
  const float* H      = (const float*)d_in[0];  // [512,4096]
  const float* p      = (const float*)d_in[1];  // [4096,4096] (symmetric)
  const float* beta   = (const float*)d_in[2];  // [4096,128]
  const float* target = (const float*)d_in[3];  // [512,128]
  float* out = (float*)d_out;                   // [512,128]

  // workspace layout (floats)
  float* ws   = (float*)d_ws;
  float* Hp   = ws;               ws += (size_t)Bz * Nh;  // 512x4096
  float* S    = ws;               ws += (size_t)Bz * Bz;  // 512x512
  float* Xa   = ws;               ws += (size_t)Bz * Bz;
  float* Xb   = ws;               ws += (size_t)Bz * Bz;
  float* Y    = ws;               ws += (size_t)Bz * Bz;
  float* G    = ws;               ws += (size_t)Nh * No;  // 4096x128
  float* pG   = ws;               ws += (size_t)Nh * No;
  float* Z    = ws;               ws += (size_t)Nh * No;
  float* bnew = ws;               ws += (size_t)Nh * No;
  float* R    = ws;               ws += (size_t)Bz * No;  // 512x128
  float* W    = ws;               ws += (size_t)Bz * No;
  float* V    = ws;               ws += (size_t)Bz * No;
  float* invn = ws;               ws += 4;

  const dim3 blk(128);
  auto grid = [](int M, int N) { return dim3(N / BN, M / BM); };

  // 1) Hp = H * p
  gemm_f32_wmma<false, false><<<grid(Bz, Nh), blk, 0, stream>>>(
      H, p, Hp, nullptr, Bz, Nh, Nh, Nh, Nh, Nh, 1.f, 0.f, 0.f);
  // 2) S = Hp * H^T + I
  gemm_f32_wmma<false, true><<<grid(Bz, Bz), blk, 0, stream>>>(
      Hp, H, S, nullptr, Bz, Bz, Nh, Nh, Nh, Bz, 1.f, 0.f, 1.f);
  // 3) alpha = 1/||S||_inf ; 4) Xa = alpha * I
  rownorm_inf_inv<<<1, 512, 0, stream>>>(S, Bz, invn);
  init_scaled_identity<<<(Bz * Bz + 255) / 256, 256, 0, stream>>>(Xa, Bz, invn);
  // 5) Newton-Schulz: X <- 2X - X*S*X  (S SPD, X0=I/||S||inf => guaranteed conv.)
  for (int it = 0; it < 16; ++it) {
    gemm_f32_wmma<false, false><<<grid(Bz, Bz), blk, 0, stream>>>(
        S, Xa, Y, nullptr, Bz, Bz, Bz, Bz, Bz, Bz, 1.f, 0.f, 0.f);
    gemm_f32_wmma<false, false><<<grid(Bz, Bz), blk, 0, stream>>>(
        Xa, Y, Xb, Xa, Bz, Bz, Bz, Bz, Bz, Bz, -1.f, 2.f, 0.f);
    float* t = Xa; Xa = Xb; Xb = t;
  }
  // Xa now holds T = inv(S)

  // 6) R = target - H*beta
  gemm_f32_wmma<false, false><<<grid(Bz, No), blk, 0, stream>>>(
      H, beta, R, target, Bz, No, Nh, Nh, No, No, -1.f, 1.f, 0.f);
  // 7) G = H^T * R
  gemm_f32_wmma<true, false><<<grid(Nh, No), blk, 0, stream>>>(
      H, R, G, nullptr, Nh, No, Bz, Nh, No, No, 1.f, 0.f, 0.f);
  // 8) pG = p * G
  gemm_f32_wmma<false, false><<<grid(Nh, No), blk, 0, stream>>>(
      p, G, pG, nullptr, Nh, No, Nh, Nh, No, No, 1.f, 0.f, 0.f);
  // 9) W = Hp * G
  gemm_f32_wmma<false, false><<<grid(Bz, No), blk, 0, stream>>>(
      Hp, G, W, nullptr, Bz, No, Nh, Nh, No, No, 1.f, 0.f, 0.f);
  // 10) V = T * W
  gemm_f32_wmma<false, false><<<grid(Bz, No), blk, 0, stream>>>(
      Xa, W, V, nullptr, Bz, No, Bz, Bz, No, No, 1.f, 0.f, 0.f);
  // 11) Z = Hp^T * V   (= p H^T T Hp G, using p symmetric)
  gemm_f32_wmma<true, false><<<grid(Nh, No), blk, 0, stream>>>(
      Hp, V, Z, nullptr, Nh, No, Bz, Nh, No, No, 1.f, 0.f, 0.f);
  // 12) bnew = beta + pG - Z
  beta_combine<<<(Nh * No + 255) / 256, 256, 0, stream>>>(beta, pG, Z, bnew,
                                                          Nh * No);
  // 13) out = H * bnew
  gemm_f32_wmma<false, false><<<grid(Bz, No), blk, 0, stream>>>(
      H, bnew, out, nullptr, Bz, No, Nh, Nh, No, No, 1.f, 0.f, 0.f);
}